// GCNRegression_63780264346286
// MI455X (gfx1250) — compile-verified
//
#include <hip/hip_runtime.h>
#include <hip/hip_bf16.h>

#define D_FEAT 256
#define HID    16
#define KSTEPS (D_FEAT / 4)   // 64 WMMA K-steps of 4
#define KCHUNK 8              // A-fragment load batch size

typedef float v2f __attribute__((ext_vector_type(2)));
typedef float v8f __attribute__((ext_vector_type(8)));

// ---------------------------------------------------------------------------
// 1) In-degree of each dst node (self-loop added later as +1).
// ---------------------------------------------------------------------------
__global__ void gcn_deg_kernel(const int* __restrict__ dst,
                               unsigned* __restrict__ deg, int E) {
    int e = blockIdx.x * blockDim.x + threadIdx.x;
    if (e < E) atomicAdd(&deg[dst[e]], 1u);
}

// ---------------------------------------------------------------------------
// 2) dinv = rsqrt(deg + 1)
// ---------------------------------------------------------------------------
__global__ void gcn_dinv_kernel(const unsigned* __restrict__ deg,
                                float* __restrict__ dinv, int N) {
    int i = blockIdx.x * blockDim.x + threadIdx.x;
    if (i < N) dinv[i] = rsqrtf((float)(deg[i] + 1u));
}

// ---------------------------------------------------------------------------
// 3) h = x @ W  via V_WMMA_F32_16X16X4_F32.
//    One wave owns one 16x16 output tile. W staged in LDS, then all 64 B
//    fragments held in registers (128 VGPRs) for the whole tile loop.
//    __launch_bounds__(256,1) expands the VGPR budget so nothing spills.
//    A fragments are loaded in batches of 8 so 8 global_load_b64 are in
//    flight per s_wait, instead of load->wait0->wmma serialization.
// ---------------------------------------------------------------------------
__global__ __launch_bounds__(256, 1)
void gcn_gemm_wmma_kernel(const float* __restrict__ x,
                          const float* __restrict__ W,
                          float* __restrict__ h, int ntiles) {
    __shared__ float Wl[D_FEAT * HID];

    const int tid = threadIdx.x;
    // Stage W into LDS (row-major: [k][n]); 4096 floats / 256 threads.
    for (int i = tid; i < (D_FEAT * HID) / 4; i += blockDim.x)
        ((float4*)Wl)[i] = ((const float4*)W)[i];
    __syncthreads();

    const int lane  = tid & 31;
    const int wave  = tid >> 5;
    const int wpb   = blockDim.x >> 5;
    const int gwave = blockIdx.x * wpb + wave;
    const int nwav  = gridDim.x * wpb;

    const int mrow = lane & 15;        // A-matrix row (M) for this lane
    const int kk   = (lane >> 4) << 1; // K sub-offset: 0 (lanes 0-15) or 2
    const int ncol = lane & 15;        // B/C-matrix column (N) for this lane

    // Preload ALL B fragments into registers: 64 x v2f = 128 VGPRs.
    // B layout (4x16 f32): lane holds W[k+kk, n] and W[k+kk+1, n].
    v2f bfrag[KSTEPS];
#pragma unroll
    for (int kb = 0; kb < KSTEPS; ++kb) {
        bfrag[kb].x = Wl[(4 * kb + kk)     * HID + ncol];
        bfrag[kb].y = Wl[(4 * kb + kk + 1) * HID + ncol];
    }

    for (int tile = gwave; tile < ntiles; tile += nwav) {
        const float* xrow = x + (size_t)(tile * 16 + mrow) * D_FEAT + kk;

        // Prefetch next tile's row for this lane (global_prefetch_b8).
        if (tile + nwav < ntiles)
            __builtin_prefetch(xrow + (size_t)nwav * 16 * D_FEAT, 0, 0);

        v8f c = {};
#pragma unroll
        for (int kc = 0; kc < KSTEPS; kc += KCHUNK) {
            v2f a[KCHUNK];
#pragma unroll
            for (int u = 0; u < KCHUNK; ++u)        // 8 loads in flight
                a[u] = *(const v2f*)(xrow + 4 * (kc + u));
#pragma unroll
            for (int u = 0; u < KCHUNK; ++u)        // then 8 WMMAs
                c = __builtin_amdgcn_wmma_f32_16x16x4_f32(
                        false, a[u], false, bfrag[kc + u],
                        (short)0, c, false, false);
        }

        // C/D layout: VGPR v -> row M = v + 8*(lane>>4), col N = lane&15.
        float* hout = h + (size_t)tile * 16 * HID + (lane >> 4) * 8 * HID + ncol;
#pragma unroll
        for (int v = 0; v < 8; ++v)
            hout[v * HID] = c[v];
    }
}

// ---------------------------------------------------------------------------
// 4) Edge propagation: agg[dst] += h[src] * dinv[src]*dinv[dst]
//    One thread per edge; fp32 HW atomics (h/agg live in the 192MB L2).
// ---------------------------------------------------------------------------
__global__ void gcn_scatter_kernel(const int* __restrict__ ei,
                                   const float* __restrict__ dinv,
                                   const float* __restrict__ h,
                                   float* __restrict__ agg, int E) {
    for (int e = blockIdx.x * blockDim.x + threadIdx.x; e < E;
         e += gridDim.x * blockDim.x) {
        const int s = ei[e];
        const int d = ei[E + e];
        const float w = dinv[s] * dinv[d];
        const float4* hs = (const float4*)(h + (size_t)s * HID);
        float* ad = agg + (size_t)d * HID;
#pragma unroll
        for (int j = 0; j < 4; ++j) {
            float4 v = hs[j];
            unsafeAtomicAdd(ad + 4 * j + 0, v.x * w);
            unsafeAtomicAdd(ad + 4 * j + 1, v.y * w);
            unsafeAtomicAdd(ad + 4 * j + 2, v.z * w);
            unsafeAtomicAdd(ad + 4 * j + 3, v.w * w);
        }
    }
}

// ---------------------------------------------------------------------------
// 5) Head: out[i] = sum_j (agg[i][j] + h[i][j]*dinv[i]^2 + b_conv[j]) * W_fc[j]
//                   + b_fc      (self-loop message folded in here)
// ---------------------------------------------------------------------------
__global__ void gcn_head_kernel(const float* __restrict__ h,
                                const float* __restrict__ agg,
                                const float* __restrict__ dinv,
                                const float* __restrict__ bconv,
                                const float* __restrict__ wfc,
                                const float* __restrict__ bfc,
                                float* __restrict__ out, int N) {
    int i = blockIdx.x * blockDim.x + threadIdx.x;
    if (i >= N) return;
    const float di = dinv[i];
    const float sw = di * di;
    const float4* hp = (const float4*)(h + (size_t)i * HID);
    const float4* ap = (const float4*)(agg + (size_t)i * HID);
    float acc = 0.f;
#pragma unroll
    for (int j = 0; j < 4; ++j) {
        float4 hv = hp[j];
        float4 av = ap[j];
        float4 wv = ((const float4*)wfc)[j];
        float4 bv = ((const float4*)bconv)[j];
        acc += (av.x + hv.x * sw + bv.x) * wv.x;
        acc += (av.y + hv.y * sw + bv.y) * wv.y;
        acc += (av.z + hv.z * sw + bv.z) * wv.z;
        acc += (av.w + hv.w * sw + bv.w) * wv.w;
    }
    out[i] = acc + bfc[0];
}

// ---------------------------------------------------------------------------
// Launch: deg -> dinv -> GEMM(WMMA) -> scatter -> head, all on `stream`.
// ---------------------------------------------------------------------------
extern "C" void kernel_launch(void* const* d_in, const int* in_sizes, int n_in,
                              void* d_out, int out_size, void* d_ws, size_t ws_size,
                              hipStream_t stream) {
    const float* x     = (const float*)d_in[0];
    const int*   ei    = (const int*)d_in[1];   // [2, E]: src row then dst row
    const float* W     = (const float*)d_in[2];
    const float* bconv = (const float*)d_in[3];
    const float* wfc   = (const float*)d_in[4];
    const float* bfc   = (const float*)d_in[5];
    float* out = (float*)d_out;

    const int N = in_sizes[0] / D_FEAT;   // 100000
    const int E = in_sizes[1] / 2;        // 3200000

    // Workspace layout (16B-aligned slices):
    char* ws = (char*)d_ws;
    unsigned* deg  = (unsigned*)ws;                                      // N u32
    float*    dinv = (float*)(ws + (size_t)N * 4);                       // N f32
    float*    h    = (float*)(ws + (size_t)N * 8);                       // N*16 f32
    float*    agg  = (float*)(ws + (size_t)N * 8 + (size_t)N * HID * 4); // N*16 f32

    hipMemsetAsync(deg, 0, (size_t)N * sizeof(unsigned), stream);
    hipMemsetAsync(agg, 0, (size_t)N * HID * sizeof(float), stream);

    gcn_deg_kernel<<<(E + 255) / 256, 256, 0, stream>>>(ei + E, deg, E);
    gcn_dinv_kernel<<<(N + 255) / 256, 256, 0, stream>>>(deg, dinv, N);

    const int ntiles = N / 16;            // N is a multiple of 16 (6250 tiles)
    gcn_gemm_wmma_kernel<<<128, 256, 0, stream>>>(x, W, h, ntiles);

    gcn_scatter_kernel<<<1024, 256, 0, stream>>>(ei, dinv, h, agg, E);

    gcn_head_kernel<<<(N + 255) / 256, 256, 0, stream>>>(h, agg, dinv, bconv,
                                                         wfc, bfc, out, N);
}